// TokenTransformer_76562087018549
// MI455X (gfx1250) — compile-verified
//
#include <hip/hip_runtime.h>
#include <stdint.h>

typedef __attribute__((ext_vector_type(2))) float v2f;
typedef __attribute__((ext_vector_type(8))) float v8f;

#define TM 64
#define TN 128
#define TK 16

// Generic batched GEMM on the fp32 WMMA path:
//   C[z][m,n] = (Cin ? Cin[z][m,n] : 0) + sum_{k1,k2} A[z][m,k1,k2] * B[z][k1,k2,n]
// z = blockIdx.z decomposed as (b = z/Hdim, h = z%Hdim); flattened k = k1*K2+k2.
// Block = 256 threads (8 wave32s) arranged 2(M) x 4(N); each wave owns a
// 32x32 accumulator footprint = 2x2 V_WMMA_F32_16X16X4_F32 tiles.
// Requires: M % 64 == 0, N % 128 == 0, K2 % 16 == 0 (true for every call here).
__global__ __launch_bounds__(256) void gemm_wmma_f32(
    const float* __restrict__ A, const float* __restrict__ B,
    const float* __restrict__ Cin, float* __restrict__ Cout,
    int Ktot, int K2, int Hdim,
    int64_t a_zb, int64_t a_zh, int64_t a_sm, int64_t a_sk1, int64_t a_sk2,
    int64_t b_zb, int64_t b_zh, int64_t b_sn, int64_t b_sk1, int64_t b_sk2,
    int64_t c_zb, int64_t c_zh, int64_t c_sm,
    int64_t ci_zb, int64_t ci_sm)
{
    __shared__ float As[TM][TK + 1];
    __shared__ float Bs[TK][TN + 1];

    const int t    = threadIdx.x;
    const int lane = t & 31;
    const int wave = t >> 5;
    const int wr   = wave >> 2;   // 0..1  (M direction, 32 rows each)
    const int wc   = wave & 3;    // 0..3  (N direction, 32 cols each)

    const int zh = blockIdx.z % Hdim;
    const int zb = blockIdx.z / Hdim;

    const float* Az = A + (int64_t)zb * a_zb + (int64_t)zh * a_zh;
    const float* Bz = B + (int64_t)zb * b_zb + (int64_t)zh * b_zh;

    const int m0 = blockIdx.y * TM;
    const int n0 = blockIdx.x * TN;

    // WMMA fragment coordinates (ISA 7.12.2):
    // A 16x4 f32: lanes 0-15 -> M=lane, K={0,1}; lanes 16-31 -> M=lane-16, K={2,3}
    const int l15 = lane & 15;
    const int klo = (lane >> 4) << 1;           // 0 or 2
    const int amr = wr * 32 + l15;              // A rows: +0 / +16 for the two M subtiles
    const int bnc = wc * 32 + l15;              // B cols: +0 / +16 for the two N subtiles

    v8f acc00 = {}, acc01 = {}, acc10 = {}, acc11 = {};

    for (int kc = 0; kc < Ktot; kc += TK) {
        const int k1  = kc / K2;                // K-chunk never crosses k1 (K2 % 16 == 0)
        const int k2b = kc - k1 * K2;

        // Stage A tile: 64x16 = 1024 floats, 4 per thread
        #pragma unroll
        for (int p = 0; p < 4; ++p) {
            int e = t + p * 256;
            int i = e >> 4;
            int j = e & 15;
            As[i][j] = Az[(int64_t)(m0 + i) * a_sm + (int64_t)k1 * a_sk1 +
                          (int64_t)(k2b + j) * a_sk2];
        }
        // Stage B tile: 16x128 = 2048 floats, 8 per thread
        #pragma unroll
        for (int p = 0; p < 8; ++p) {
            int e = t + p * 256;
            int n = e & 127;
            int j = e >> 7;
            Bs[j][n] = Bz[(int64_t)k1 * b_sk1 + (int64_t)(k2b + j) * b_sk2 +
                          (int64_t)(n0 + n) * b_sn];
        }
        __syncthreads();

        #pragma unroll
        for (int kk = 0; kk < TK; kk += 4) {
            v2f a0, a1, b0, b1;
            a0[0] = As[amr][kk + klo];
            a0[1] = As[amr][kk + klo + 1];
            a1[0] = As[amr + 16][kk + klo];
            a1[1] = As[amr + 16][kk + klo + 1];
            b0[0] = Bs[kk + klo][bnc];
            b0[1] = Bs[kk + klo + 1][bnc];
            b1[0] = Bs[kk + klo][bnc + 16];
            b1[1] = Bs[kk + klo + 1][bnc + 16];
            acc00 = __builtin_amdgcn_wmma_f32_16x16x4_f32(false, a0, false, b0,
                        (short)0, acc00, false, false);
            acc01 = __builtin_amdgcn_wmma_f32_16x16x4_f32(false, a0, false, b1,
                        (short)0, acc01, false, false);
            acc10 = __builtin_amdgcn_wmma_f32_16x16x4_f32(false, a1, false, b0,
                        (short)0, acc10, false, false);
            acc11 = __builtin_amdgcn_wmma_f32_16x16x4_f32(false, a1, false, b1,
                        (short)0, acc11, false, false);
        }
        __syncthreads();
    }

    // C/D layout: VGPR r -> lanes 0-15: M=base+r, N=lane; lanes 16-31: M=base+8+r, N=lane-16
    const int mhalf = (lane >> 4) << 3;         // 0 or 8
    const int64_t cz = (int64_t)zb * c_zb + (int64_t)zh * c_zh;
    v8f accs[4] = { acc00, acc01, acc10, acc11 };
    const float* Ciz = Cin ? (Cin + (int64_t)zb * ci_zb) : nullptr;

    #pragma unroll
    for (int mi = 0; mi < 2; ++mi) {
        #pragma unroll
        for (int ni = 0; ni < 2; ++ni) {
            const v8f& a = accs[mi * 2 + ni];
            const int mb = m0 + wr * 32 + mi * 16 + mhalf;
            const int nc = n0 + wc * 32 + ni * 16 + l15;
            if (Ciz) {
                #pragma unroll
                for (int r = 0; r < 8; ++r)
                    Cout[cz + (int64_t)(mb + r) * c_sm + nc] =
                        a[r] + Ciz[(int64_t)(mb + r) * ci_sm + nc];
            } else {
                #pragma unroll
                for (int r = 0; r < 8; ++r)
                    Cout[cz + (int64_t)(mb + r) * c_sm + nc] = a[r];
            }
        }
    }
}

// In-place softmax over rows of length S=512 with causal mask + 1/sqrt(P) scale.
// One block per (b,h,q) row; 256 threads x 2 elements.
__global__ __launch_bounds__(256) void softmax_rows(
    float* __restrict__ att, const float* __restrict__ mask,
    float inv_scale, int S)
{
    const int64_t row = blockIdx.x;
    const int q = (int)(row % S);
    float* p = att + row * (int64_t)S;
    const float* mrow = mask + (int64_t)q * S;
    const int t = threadIdx.x;

    float x0 = p[t]       * inv_scale + mrow[t];
    float x1 = p[t + 256] * inv_scale + mrow[t + 256];

    __shared__ float red[256];
    red[t] = fmaxf(x0, x1);
    __syncthreads();
    for (int off = 128; off > 0; off >>= 1) {
        if (t < off) red[t] = fmaxf(red[t], red[t + off]);
        __syncthreads();
    }
    const float rmax = red[0];
    __syncthreads();

    const float e0 = __expf(x0 - rmax);
    const float e1 = __expf(x1 - rmax);
    red[t] = e0 + e1;
    __syncthreads();
    for (int off = 128; off > 0; off >>= 1) {
        if (t < off) red[t] += red[t + off];
        __syncthreads();
    }
    const float inv = 1.0f / red[0];

    p[t]       = e0 * inv;
    p[t + 256] = e1 * inv;
}

extern "C" void kernel_launch(void* const* d_in, const int* in_sizes, int n_in,
                              void* d_out, int out_size, void* d_ws, size_t ws_size,
                              hipStream_t stream)
{
    const int Bb = 8, S = 512, V = 256, D = 1024, P = 128, H = 8, L = 4;

    const float* tokens    = (const float*)d_in[0];
    const float* att_mask  = (const float*)d_in[1];
    const float* embed_W   = (const float*)d_in[2];
    const float* unembed_W = (const float*)d_in[3];
    const float* wq        = (const float*)d_in[4];
    const float* wk        = (const float*)d_in[5];
    const float* wv        = (const float*)d_in[6];
    const float* wo        = (const float*)d_in[7];

    // Output layout: logits [B,S,V] | atts [L,B,H,S,S] | streams [L+1,B,S,D]
    float* out     = (float*)d_out;
    float* logits  = out;
    float* atts    = out + (int64_t)Bb * S * V;
    float* streams = atts + (int64_t)L * Bb * H * S * S;

    // Workspace: q, k, v, comb — each [B,H,S,P] = 16 MB (64 MB total)
    const int64_t QKV = (int64_t)Bb * H * S * P;
    float* qb = (float*)d_ws;
    float* kb = qb + QKV;
    float* vb = kb + QKV;
    float* cb = vb + QKV;

    const dim3 blk(256);
    const float inv_scale = 0.08838834764831845f;  // 1/sqrt(128)

    // ---- embed: stream0[b*S+s, d] = tokens @ embed_W ----
    gemm_wmma_f32<<<dim3(D / TN, (Bb * S) / TM, 1), blk, 0, stream>>>(
        tokens, embed_W, nullptr, streams,
        /*Ktot=*/V, /*K2=*/V, /*Hdim=*/1,
        /*A*/ 0, 0, V, 0, 1,
        /*B*/ 0, 0, 1, 0, D,
        /*C*/ 0, 0, D,
        /*Ci*/ 0, 0);

    for (int l = 0; l < L; ++l) {
        const float* st_in  = streams + (int64_t)l * Bb * S * D;
        float*       st_out = streams + (int64_t)(l + 1) * Bb * S * D;
        float*       att_l  = atts + (int64_t)l * Bb * H * S * S;
        const float* wql = wq + (int64_t)l * H * D * P;
        const float* wkl = wk + (int64_t)l * H * D * P;
        const float* wvl = wv + (int64_t)l * H * D * P;
        const float* wol = wo + (int64_t)l * H * P * D;

        // ---- q/k/v projections: [B*H] x (S x D x P) ----
        const float* wts[3] = { wql, wkl, wvl };
        float*       dst[3] = { qb, kb, vb };
        for (int i = 0; i < 3; ++i) {
            gemm_wmma_f32<<<dim3(P / TN, S / TM, Bb * H), blk, 0, stream>>>(
                st_in, wts[i], nullptr, dst[i],
                /*Ktot=*/D, /*K2=*/D, /*Hdim=*/H,
                /*A*/ (int64_t)S * D, 0, D, 0, 1,
                /*B*/ 0, (int64_t)D * P, 1, 0, P,
                /*C*/ (int64_t)H * S * P, (int64_t)S * P, P,
                /*Ci*/ 0, 0);
        }

        // ---- logits = q @ k^T : [B*H] x (S x P x S), written raw into atts ----
        gemm_wmma_f32<<<dim3(S / TN, S / TM, Bb * H), blk, 0, stream>>>(
            qb, kb, nullptr, att_l,
            /*Ktot=*/P, /*K2=*/P, /*Hdim=*/H,
            /*A*/ (int64_t)H * S * P, (int64_t)S * P, P, 0, 1,
            /*B*/ (int64_t)H * S * P, (int64_t)S * P, /*b_sn=*/P, 0, /*b_sk2=*/1,
            /*C*/ (int64_t)H * S * S, (int64_t)S * S, S,
            /*Ci*/ 0, 0);

        // ---- softmax (scale + causal mask fused), in place in atts ----
        softmax_rows<<<dim3(Bb * H * S), blk, 0, stream>>>(att_l, att_mask, inv_scale, S);

        // ---- comb = att @ v : [B*H] x (S x S x P) ----
        gemm_wmma_f32<<<dim3(P / TN, S / TM, Bb * H), blk, 0, stream>>>(
            att_l, vb, nullptr, cb,
            /*Ktot=*/S, /*K2=*/S, /*Hdim=*/H,
            /*A*/ (int64_t)H * S * S, (int64_t)S * S, S, 0, 1,
            /*B*/ (int64_t)H * S * P, (int64_t)S * P, 1, 0, P,
            /*C*/ (int64_t)H * S * P, (int64_t)S * P, P,
            /*Ci*/ 0, 0);

        // ---- stream_{l+1} = stream_l + sum_h comb[b,h] @ wo[h]  (K = H*P, two-level) ----
        gemm_wmma_f32<<<dim3(D / TN, S / TM, Bb), blk, 0, stream>>>(
            cb, wol, st_in, st_out,
            /*Ktot=*/H * P, /*K2=*/P, /*Hdim=*/1,
            /*A*/ (int64_t)H * S * P, 0, P, (int64_t)S * P, 1,
            /*B*/ 0, 0, 1, (int64_t)P * D, D,
            /*C*/ (int64_t)S * D, 0, D,
            /*Ci*/ (int64_t)S * D, D);
    }

    // ---- unembed: logits = stream_L @ unembed_W ----
    gemm_wmma_f32<<<dim3(V / TN, (Bb * S) / TM, 1), blk, 0, stream>>>(
        streams + (int64_t)L * Bb * S * D, unembed_W, nullptr, logits,
        /*Ktot=*/D, /*K2=*/D, /*Hdim=*/1,
        /*A*/ 0, 0, D, 0, 1,
        /*B*/ 0, 0, 1, 0, V,
        /*C*/ 0, 0, V,
        /*Ci*/ 0, 0);
}